// EnhancedEntityNBFNet_71227737637026
// MI455X (gfx1250) — compile-verified
//
#include <hip/hip_runtime.h>

#define N_NODES 50000
#define E_EDGES 800000
#define R_REL   64
#define D_DIM   64
#define L_LAYERS 4
#define B_BATCH 2
#define K_TAILS 32

typedef float v2f __attribute__((ext_vector_type(2)));
typedef float v8f __attribute__((ext_vector_type(8)));

// ---------- init: boundary = boundary_extra ; x = boundary ----------
__global__ void init_copy4(const float4* __restrict__ be,
                           float4* __restrict__ boundary,
                           float4* __restrict__ x, int n4) {
    int i = blockIdx.x * blockDim.x + threadIdx.x;
    int stride = gridDim.x * blockDim.x;
    for (; i < n4; i += stride) { float4 v = be[i]; boundary[i] = v; x[i] = v; }
}

// ---------- add query at h_index (128 threads: b*64 + d) ----------
__global__ void add_query(const int* __restrict__ h_index,
                          const int* __restrict__ r_index,
                          const float* __restrict__ rel,
                          float* __restrict__ boundary,
                          float* __restrict__ x) {
    int t = threadIdx.x;
    int b = t >> 6, d = t & 63;
    float q = rel[((size_t)b * R_REL + r_index[b]) * D_DIM + d];
    size_t off = ((size_t)b * N_NODES + h_index[b]) * D_DIM + d;
    boundary[off] += q;
    x[off] += q;
}

// ---------- agg = boundary (per layer) ----------
__global__ void copy4(const float4* __restrict__ srcv, float4* __restrict__ dstv, int n4) {
    int i = blockIdx.x * blockDim.x + threadIdx.x;
    int stride = gridDim.x * blockDim.x;
    for (; i < n4; i += stride) dstv[i] = srcv[i];
}

// ---------- edge scatter: agg[b,dst,:] += x[b,src,:] * rel[b,etype,:] ----------
__global__ void scatter_msgs(const int* __restrict__ src, const int* __restrict__ dst,
                             const int* __restrict__ etype, const float* __restrict__ rel,
                             const float* __restrict__ x, float* __restrict__ agg) {
    long long t = (long long)blockIdx.x * blockDim.x + threadIdx.x;
    if (t >= (long long)E_EDGES * D_DIM) return;
    int e = (int)(t >> 6);
    int d = (int)(t & 63);
    int s = src[e], dd = dst[e], r = etype[e];
#pragma unroll
    for (int b = 0; b < B_BATCH; ++b) {
        float xv = x[((size_t)b * N_NODES + s) * D_DIM + d];
        float rv = rel[((size_t)b * R_REL + r) * D_DIM + d];
        atomicAdd(&agg[((size_t)b * N_NODES + dd) * D_DIM + d], xv * rv);
    }
}

// ---------- per-layer: out = relu(LN(concat(x,agg) @ W + b)) ; x += out ----------
// One wave32 per 16-row tile. K = 128 via 32 x V_WMMA_F32_16X16X4_F32 per F-tile.
__global__ __launch_bounds__(256) void layer_gemm(
    float* __restrict__ x, const float* __restrict__ agg,
    const float* __restrict__ W, const float* __restrict__ bias,
    const float* __restrict__ g, const float* __restrict__ bb) {
    const int lane = threadIdx.x & 31;
    const int wave = threadIdx.x >> 5;
    const int tile = blockIdx.x * 8 + wave;
    const int NT = (B_BATCH * N_NODES) / 16;   // 6250, batch boundary tile-aligned
    if (tile >= NT) return;

    const int half = lane >> 4;      // 0: K rows {0,1}, 1: K rows {2,3}
    const int lm   = lane & 15;      // A: row M ; B/C: column N
    const size_t row0 = (size_t)tile * 16;
    const float* arow  = x   + (row0 + lm) * D_DIM;   // K  0..63
    const float* arow2 = agg + (row0 + lm) * D_DIM;   // K 64..127

    v8f acc0 = {}, acc1 = {}, acc2 = {}, acc3 = {};

#pragma unroll
    for (int kc = 0; kc < 32; ++kc) {
        // A tile 16x4: lane loads (M=lm, K=kbase+2*half .. +1) as one 8B load
        const float* ap = (kc < 16) ? (arow  + kc * 4 + 2 * half)
                                    : (arow2 + (kc - 16) * 4 + 2 * half);
        v2f a = *(const v2f*)ap;

        // B tile 4x16 per F-tile: lane holds column f = ft*16+lm, K rows 2*half,2*half+1
        const float* wp = W + (size_t)(kc * 4 + 2 * half) * D_DIM + lm;
        v2f b0; b0.x = wp[0];  b0.y = wp[64];
        v2f b1; b1.x = wp[16]; b1.y = wp[80];
        v2f b2; b2.x = wp[32]; b2.y = wp[96];
        v2f b3; b3.x = wp[48]; b3.y = wp[112];

        acc0 = __builtin_amdgcn_wmma_f32_16x16x4_f32(false, a, false, b0, (short)0, acc0, false, false);
        acc1 = __builtin_amdgcn_wmma_f32_16x16x4_f32(false, a, false, b1, (short)0, acc1, false, false);
        acc2 = __builtin_amdgcn_wmma_f32_16x16x4_f32(false, a, false, b2, (short)0, acc2, false, false);
        acc3 = __builtin_amdgcn_wmma_f32_16x16x4_f32(false, a, false, b3, (short)0, acc3, false, false);
    }

    // Fused bias + LayerNorm + ReLU + residual.
    // C layout: VGPR j holds row (j + 8*half), column lm of each 16-wide F-tile.
    const float inv64 = 1.0f / 64.0f;
    const float bi0 = bias[lm], bi1 = bias[16 + lm], bi2 = bias[32 + lm], bi3 = bias[48 + lm];
    const float g0 = g[lm],  g1 = g[16 + lm],  g2 = g[32 + lm],  g3 = g[48 + lm];
    const float c0 = bb[lm], c1 = bb[16 + lm], c2 = bb[32 + lm], c3 = bb[48 + lm];

#pragma unroll
    for (int j = 0; j < 8; ++j) {
        float v0 = acc0[j] + bi0, v1 = acc1[j] + bi1, v2 = acc2[j] + bi2, v3 = acc3[j] + bi3;
        float s = v0 + v1 + v2 + v3;
        s += __shfl_xor(s, 1, 32); s += __shfl_xor(s, 2, 32);
        s += __shfl_xor(s, 4, 32); s += __shfl_xor(s, 8, 32);
        const float mu = s * inv64;
        v0 -= mu; v1 -= mu; v2 -= mu; v3 -= mu;
        float q = v0 * v0 + v1 * v1 + v2 * v2 + v3 * v3;
        q += __shfl_xor(q, 1, 32); q += __shfl_xor(q, 2, 32);
        q += __shfl_xor(q, 4, 32); q += __shfl_xor(q, 8, 32);
        const float rstd = rsqrtf(q * inv64 + 1e-5f);

        const int rj = j + 8 * half;
        float* xr = x + (row0 + rj) * D_DIM + lm;
        float o;
        o = v0 * rstd * g0 + c0; o = fmaxf(o, 0.0f); xr[0]  += o;
        o = v1 * rstd * g1 + c1; o = fmaxf(o, 0.0f); xr[16] += o;
        o = v2 * rstd * g2 + c2; o = fmaxf(o, 0.0f); xr[32] += o;
        o = v3 * rstd * g3 + c3; o = fmaxf(o, 0.0f); xr[48] += o;
    }
}

// ---------- final MLP scoring: 64 (b,k) pairs ----------
__global__ void score_kernel(const float* __restrict__ x, const int* __restrict__ t_index,
                             const int* __restrict__ r_index, const float* __restrict__ rel,
                             const float* __restrict__ w1, const float* __restrict__ b1,
                             const float* __restrict__ w2, const float* __restrict__ b2,
                             float* __restrict__ out) {
    int t = threadIdx.x;
    if (t >= B_BATCH * K_TAILS) return;
    int b = t >> 5, k = t & 31;
    int n = t_index[b * K_TAILS + k];
    const float* xf = x   + ((size_t)b * N_NODES + n) * D_DIM;
    const float* qf = rel + ((size_t)b * R_REL + r_index[b]) * D_DIM;
    float score = b2[0];
    for (int f = 0; f < D_DIM; ++f) {
        float h = b1[f];
        for (int j = 0; j < D_DIM; ++j) h += xf[j] * w1[j * D_DIM + f];
        for (int j = 0; j < D_DIM; ++j) h += qf[j] * w1[(D_DIM + j) * D_DIM + f];
        h = fmaxf(h, 0.0f);
        score += h * w2[f];
    }
    out[t] = score;
}

extern "C" void kernel_launch(void* const* d_in, const int* in_sizes, int n_in,
                              void* d_out, int out_size, void* d_ws, size_t ws_size,
                              hipStream_t stream) {
    const int*   edge_index = (const int*)d_in[0];
    const int*   src        = edge_index;
    const int*   dst        = edge_index + E_EDGES;
    const int*   edge_type  = (const int*)d_in[1];
    const int*   h_index    = (const int*)d_in[2];
    const int*   r_index    = (const int*)d_in[3];
    const int*   t_index    = (const int*)d_in[4];
    const float* rel        = (const float*)d_in[5];
    const float* be         = (const float*)d_in[6];
    const float* layers_W   = (const float*)d_in[7];
    const float* layers_b   = (const float*)d_in[8];
    const float* ln_g       = (const float*)d_in[9];
    const float* ln_b       = (const float*)d_in[10];
    const float* w1         = (const float*)d_in[11];
    const float* b1         = (const float*)d_in[12];
    const float* w2         = (const float*)d_in[13];
    const float* b2         = (const float*)d_in[14];

    const size_t SZ = (size_t)B_BATCH * N_NODES * D_DIM;  // 6.4M floats
    float* ws       = (float*)d_ws;
    float* boundary = ws;
    float* x        = ws + SZ;
    float* agg      = ws + 2 * SZ;

    const int n4 = (int)(SZ / 4);
    init_copy4<<<(n4 + 255) / 256, 256, 0, stream>>>((const float4*)be, (float4*)boundary,
                                                     (float4*)x, n4);
    add_query<<<1, 128, 0, stream>>>(h_index, r_index, rel, boundary, x);

    const long long nscatter = (long long)E_EDGES * D_DIM;
    const int scatter_blocks = (int)((nscatter + 255) / 256);
    const int ntile = (B_BATCH * N_NODES) / 16;

    for (int l = 0; l < L_LAYERS; ++l) {
        copy4<<<(n4 + 255) / 256, 256, 0, stream>>>((const float4*)boundary, (float4*)agg, n4);
        scatter_msgs<<<scatter_blocks, 256, 0, stream>>>(src, dst, edge_type, rel, x, agg);
        layer_gemm<<<(ntile + 7) / 8, 256, 0, stream>>>(
            x, agg,
            layers_W + (size_t)l * 2 * D_DIM * D_DIM,
            layers_b + (size_t)l * D_DIM,
            ln_g + (size_t)l * D_DIM,
            ln_b + (size_t)l * D_DIM);
    }

    score_kernel<<<1, 64, 0, stream>>>(x, t_index, r_index, rel, w1, b1, w2, b2, (float*)d_out);
}